// ROIRotate_34368328302930
// MI455X (gfx1250) — compile-verified
//
#include <hip/hip_runtime.h>
#include <hip/hip_bf16.h>

// Rotated ROI crop-and-resize with bilinear sampling (MI455X / gfx1250).
//
// Roofline: ~50MB written + L2-resident gathers (whole image set ~50MB << 192MB
// L2) => latency/bandwidth bound, ~0.3 GFLOP total. No matmul structure =>
// WMMA not applicable; TDM staging not warranted (L2-resident working set).
//
// Fast path (mw==64, HEIGHT==8, 256 thr/block): each thread owns 2 pixels with
// identical jj (rows ii and ii+4). All 8 texel-triple loads are issued
// back-to-back for maximum memory-level parallelism, then one wait, two
// blends, two coalesced 12B stores. Box params are block-uniform -> s_load.
// Generic path keeps a strided loop with global_prefetch_b8 pipelining.

#define ROI_HEIGHT 8
#define IMG_W 512
#define IMG_H 512
#define IMG_C 3
#define BLK 256

__device__ __forceinline__ float clampf(float v, float lo, float hi) {
    return fminf(fmaxf(v, lo), hi);
}

// ---------------------------------------------------------------------------
// Fast path: PIX == 2*BLK and mw power-of-two (the reference configuration:
// mw = 64 -> jj = tid & 63, ii0 = tid >> 6 in 0..3, ii1 = ii0 + 4).
// ---------------------------------------------------------------------------
__global__ __launch_bounds__(BLK) void ROIRotate_fast_kernel(
    const float* __restrict__ images,      // [N, IMG_H, IMG_W, IMG_C]
    const float* __restrict__ boxes,       // [M, 5]
    const int*   __restrict__ box_indices, // [M]
    float*       __restrict__ crops,       // [M, ROI_HEIGHT, mw, IMG_C]
    float*       __restrict__ widths_out,  // [M]
    int mw, int mw_log2)
{
    const int box = blockIdx.x;

    // Block-uniform box parameters -> scalar loads.
    const float* bx = boxes + (size_t)box * 5;
    const float x1 = bx[0], y1 = bx[1], x2 = bx[2], y2 = bx[3], ang = bx[4];
    const float bwd   = x2 - x1;
    const float bht   = y2 - y1;
    const float width = (float)ROI_HEIGHT * bwd / bht;
    const float cx    = (x1 + x2) * 0.5f;
    const float cy    = (y1 + y2) * 0.5f;
    const float s     = bht * (1.0f / (float)ROI_HEIGHT);
    const float ca = __cosf(ang);   // |ang| <= 0.785: V_COS/V_SIN accurate
    const float sa = __sinf(ang);
    const float wceil  = ceilf(width);
    const float half_w = (width - 1.0f) * 0.5f;

    if (threadIdx.x == 0) widths_out[box] = (float)mw - width;

    const int   bidx = box_indices[box];
    const float* img = images + (size_t)bidx * ((size_t)IMG_H * IMG_W * IMG_C);

    const int tid = threadIdx.x;
    const int jj  = tid & (mw - 1);
    const int ii0 = tid >> mw_log2;                 // 0 .. HEIGHT/2-1
    // pixel 1 = same jj, row ii0 + HEIGHT/2

    // Shared-jj coordinate math: dx identical for both pixels.
    const float dx  = s * ((float)jj - half_w);
    const float dy0 = s * ((float)ii0 - (float)(ROI_HEIGHT - 1) * 0.5f);
    const float dy1 = dy0 + s * (float)(ROI_HEIGHT / 2);
    const float cadx = ca * dx, sadx = sa * dx;
    const float sx0 = cx + cadx - sa * dy0;
    const float sy0 = cy + sadx + ca * dy0;
    const float sx1 = cx + cadx - sa * dy1;
    const float sy1 = cy + sadx + ca * dy1;

    // Pixel 0 indices/weights
    const float x0f0 = floorf(sx0), y0f0 = floorf(sy0);
    const float fx0 = sx0 - x0f0,   fy0 = sy0 - y0f0;
    const int ax0 = (int)clampf(x0f0,        0.0f, (float)(IMG_W - 1));
    const int bx0 = (int)clampf(x0f0 + 1.0f, 0.0f, (float)(IMG_W - 1));
    const int ay0 = (int)clampf(y0f0,        0.0f, (float)(IMG_H - 1));
    const int by0 = (int)clampf(y0f0 + 1.0f, 0.0f, (float)(IMG_H - 1));
    // Pixel 1 indices/weights
    const float x0f1 = floorf(sx1), y0f1 = floorf(sy1);
    const float fx1 = sx1 - x0f1,   fy1 = sy1 - y0f1;
    const int ax1 = (int)clampf(x0f1,        0.0f, (float)(IMG_W - 1));
    const int bx1 = (int)clampf(x0f1 + 1.0f, 0.0f, (float)(IMG_W - 1));
    const int ay1 = (int)clampf(y0f1,        0.0f, (float)(IMG_H - 1));
    const int by1 = (int)clampf(y0f1 + 1.0f, 0.0f, (float)(IMG_H - 1));

    // Texel pointers (each -> 3 adjacent floats, merges to b96 loads).
    const float* q00 = img + ((size_t)ay0 * IMG_W + ax0) * IMG_C;
    const float* q01 = img + ((size_t)ay0 * IMG_W + bx0) * IMG_C;
    const float* q10 = img + ((size_t)by0 * IMG_W + ax0) * IMG_C;
    const float* q11 = img + ((size_t)by0 * IMG_W + bx0) * IMG_C;
    const float* r00 = img + ((size_t)ay1 * IMG_W + ax1) * IMG_C;
    const float* r01 = img + ((size_t)ay1 * IMG_W + bx1) * IMG_C;
    const float* r10 = img + ((size_t)by1 * IMG_W + ax1) * IMG_C;
    const float* r11 = img + ((size_t)by1 * IMG_W + bx1) * IMG_C;

    // Issue ALL 24 channel loads (8 b96) before any use -> 8 outstanding
    // global loads per thread; single s_wait_loadcnt before the blends.
    float a0=q00[0],a1=q00[1],a2=q00[2];
    float b0=q01[0],b1=q01[1],b2=q01[2];
    float c0=q10[0],c1=q10[1],c2=q10[2];
    float d0=q11[0],d1=q11[1],d2=q11[2];
    float e0=r00[0],e1=r00[1],e2=r00[2];
    float f0=r01[0],f1=r01[1],f2=r01[2];
    float g0=r10[0],g1=r10[1],g2=r10[2];
    float h0=r11[0],h1=r11[1],h2=r11[2];

    // Weights
    const float w11a = fx0 * fy0;
    const float w01a = fx0 - w11a;
    const float w10a = fy0 - w11a;
    const float w00a = 1.0f - fx0 - fy0 + w11a;
    const float w11b = fx1 * fy1;
    const float w01b = fx1 - w11b;
    const float w10b = fy1 - w11b;
    const float w00b = 1.0f - fx1 - fy1 + w11b;

    const bool jok = ((float)jj < wceil);
    const float m0 = (jok && sx0 >= 0.0f && sx0 <= (float)(IMG_W - 1)
                          && sy0 >= 0.0f && sy0 <= (float)(IMG_H - 1)) ? 1.0f : 0.0f;
    const float m1 = (jok && sx1 >= 0.0f && sx1 <= (float)(IMG_W - 1)
                          && sy1 >= 0.0f && sy1 <= (float)(IMG_H - 1)) ? 1.0f : 0.0f;

    const int PIX = ROI_HEIGHT * mw;                 // == 2*BLK here
    float* op0 = crops + ((size_t)box * PIX + (size_t)tid) * IMG_C;
    float* op1 = op0 + (size_t)BLK * IMG_C;

    op0[0] = (a0*w00a + b0*w01a + c0*w10a + d0*w11a) * m0;
    op0[1] = (a1*w00a + b1*w01a + c1*w10a + d1*w11a) * m0;
    op0[2] = (a2*w00a + b2*w01a + c2*w10a + d2*w11a) * m0;
    op1[0] = (e0*w00b + f0*w01b + g0*w10b + h0*w11b) * m1;
    op1[1] = (e1*w00b + f1*w01b + g1*w10b + h1*w11b) * m1;
    op1[2] = (e2*w00b + f2*w01b + g2*w10b + h2*w11b) * m1;
}

// ---------------------------------------------------------------------------
// Generic fallback: any mw. Strided pixel loop with global_prefetch_b8
// pipelining of the next iteration's texel rows.
// ---------------------------------------------------------------------------
__global__ __launch_bounds__(BLK) void ROIRotate_generic_kernel(
    const float* __restrict__ images,
    const float* __restrict__ boxes,
    const int*   __restrict__ box_indices,
    float*       __restrict__ crops,
    float*       __restrict__ widths_out,
    int mw)
{
    const int box = blockIdx.x;
    const float* bx = boxes + (size_t)box * 5;
    const float x1 = bx[0], y1 = bx[1], x2 = bx[2], y2 = bx[3], ang = bx[4];
    const float bwd   = x2 - x1;
    const float bht   = y2 - y1;
    const float width = (float)ROI_HEIGHT * bwd / bht;
    const float cx    = (x1 + x2) * 0.5f;
    const float cy    = (y1 + y2) * 0.5f;
    const float s     = bht * (1.0f / (float)ROI_HEIGHT);
    const float ca = __cosf(ang);
    const float sa = __sinf(ang);
    const float wceil  = ceilf(width);
    const float half_w = (width - 1.0f) * 0.5f;

    if (threadIdx.x == 0) widths_out[box] = (float)mw - width;

    const int   bidx = box_indices[box];
    const float* img = images + (size_t)bidx * ((size_t)IMG_H * IMG_W * IMG_C);

    const int PIX = ROI_HEIGHT * mw;
    for (int p = threadIdx.x; p < PIX; p += BLK) {
        const int pn = p + BLK;
        if (pn < PIX) {
            const int   iin = pn / mw;
            const int   jjn = pn - iin * mw;
            const float dxn = s * ((float)jjn - half_w);
            const float dyn = s * ((float)iin - (float)(ROI_HEIGHT - 1) * 0.5f);
            const float sxn = cx + ca * dxn - sa * dyn;
            const float syn = cy + sa * dxn + ca * dyn;
            const int   xn  = (int)clampf(floorf(sxn),        0.0f, (float)(IMG_W - 1));
            const int   y0n = (int)clampf(floorf(syn),        0.0f, (float)(IMG_H - 1));
            const int   y1n = (int)clampf(floorf(syn) + 1.0f, 0.0f, (float)(IMG_H - 1));
            __builtin_prefetch(img + ((size_t)y0n * IMG_W + (size_t)xn) * IMG_C, 0, 3);
            __builtin_prefetch(img + ((size_t)y1n * IMG_W + (size_t)xn) * IMG_C, 0, 3);
        }

        const int   ii = p / mw;
        const int   jj = p - ii * mw;
        const float dx = s * ((float)jj - half_w);
        const float dy = s * ((float)ii - (float)(ROI_HEIGHT - 1) * 0.5f);
        const float sx = cx + ca * dx - sa * dy;
        const float sy = cy + sa * dx + ca * dy;

        const float x0f = floorf(sx), y0f = floorf(sy);
        const float fx  = sx - x0f,   fy  = sy - y0f;
        const int x0i = (int)clampf(x0f,        0.0f, (float)(IMG_W - 1));
        const int x1i = (int)clampf(x0f + 1.0f, 0.0f, (float)(IMG_W - 1));
        const int y0i = (int)clampf(y0f,        0.0f, (float)(IMG_H - 1));
        const int y1i = (int)clampf(y0f + 1.0f, 0.0f, (float)(IMG_H - 1));

        const float* r0  = img + (size_t)y0i * (IMG_W * IMG_C);
        const float* r1  = img + (size_t)y1i * (IMG_W * IMG_C);
        const float* p00 = r0 + x0i * IMG_C;
        const float* p01 = r0 + x1i * IMG_C;
        const float* p10 = r1 + x0i * IMG_C;
        const float* p11 = r1 + x1i * IMG_C;

        const float w11 = fx * fy;
        const float w01 = fx - w11;
        const float w10 = fy - w11;
        const float w00 = 1.0f - fx - fy + w11;

        const bool valid = ((float)jj < wceil)
                         && (sx >= 0.0f) && (sx <= (float)(IMG_W - 1))
                         && (sy >= 0.0f) && (sy <= (float)(IMG_H - 1));
        const float m = valid ? 1.0f : 0.0f;

        float* op = crops + ((size_t)box * PIX + (size_t)p) * IMG_C;
        op[0] = (p00[0]*w00 + p01[0]*w01 + p10[0]*w10 + p11[0]*w11) * m;
        op[1] = (p00[1]*w00 + p01[1]*w01 + p10[1]*w10 + p11[1]*w11) * m;
        op[2] = (p00[2]*w00 + p01[2]*w01 + p10[2]*w10 + p11[2]*w11) * m;
    }
}

extern "C" void kernel_launch(void* const* d_in, const int* in_sizes, int n_in,
                              void* d_out, int out_size, void* d_ws, size_t ws_size,
                              hipStream_t stream) {
    // Inputs: images, feature_map (unused by reference), boxes, box_indices,
    // max_width (device scalar; recovered on host from out_size instead).
    const float* images      = (const float*)d_in[0];
    const float* boxes       = (const float*)d_in[2];
    const int*   box_indices = (const int*)d_in[3];

    const int M  = in_sizes[3];
    const int mw = (out_size - M) / (M * ROI_HEIGHT * IMG_C);

    float* crops  = (float*)d_out;
    float* widths = (float*)d_out + (size_t)M * ROI_HEIGHT * mw * IMG_C;

    const bool pow2 = (mw > 0) && ((mw & (mw - 1)) == 0);
    if (pow2 && ROI_HEIGHT * mw == 2 * BLK) {
        int lg = 0;
        while ((1 << lg) < mw) ++lg;
        ROIRotate_fast_kernel<<<M, BLK, 0, stream>>>(
            images, boxes, box_indices, crops, widths, mw, lg);
    } else {
        ROIRotate_generic_kernel<<<M, BLK, 0, stream>>>(
            images, boxes, box_indices, crops, widths, mw);
    }
}